// SafeNpuMaxPool2d_25598005084659
// MI455X (gfx1250) — compile-verified
//
#include <hip/hip_runtime.h>

// MaxPool2d(kernel=3, stride=2, pad=1) on fp32 NCHW (16,128,224,224) -> (16,128,112,112)
// Bandwidth-bound: ~514 MB mandatory traffic => ~22 us floor at 23.3 TB/s.
//
// Each thread computes a 2x4 output tile (2 output rows x 4 output cols):
//   - needs input rows 4p-1 .. 4p+3 (middle row shared by both output rows,
//     reused in registers), cols 8q-1 .. 8q+7.
//   - per input row: 1x global_load_b32 (left col) + 2x aligned global_load_b128
//     = exactly the 9 needed columns. 15 loads + 2 NT b128 stores per 8 outputs.
//   - all padding handled by INDEX CLAMPING: max is idempotent, so clamping
//     row -1 -> 0 and col -1 -> 0 duplicates an in-window element and never
//     changes the result. Fully branchless straight-line body.

typedef __attribute__((ext_vector_type(4))) float f4;

constexpr int N_IMG = 16 * 128;          // N*C independent planes
constexpr int H = 224, W = 224;
constexpr int OH = 112, OW = 112;
constexpr unsigned QPR  = OW / 4;        // 28 column-quads per output row
constexpr unsigned PAIRS = OH / 2;       // 56 output-row pairs per image
constexpr unsigned TOTAL_THREADS = (unsigned)N_IMG * PAIRS * QPR; // 3,211,264
constexpr unsigned BLOCK = 256;          // 8 wave32 waves

static __device__ __forceinline__ f4 vmax4(f4 a, f4 b) {
    f4 r;
    r.x = __builtin_fmaxf(a.x, b.x);
    r.y = __builtin_fmaxf(a.y, b.y);
    r.z = __builtin_fmaxf(a.z, b.z);
    r.w = __builtin_fmaxf(a.w, b.w);
    return r;
}
static __device__ __forceinline__ f4 vmax3(f4 a, f4 b, f4 c) {
    return vmax4(vmax4(a, b), c);
}
static __device__ __forceinline__ float fmax3(float a, float b, float c) {
    // folds to v_max3_num_f32
    return __builtin_fmaxf(__builtin_fmaxf(a, b), c);
}

__global__ __launch_bounds__(BLOCK)
void maxpool2d_k3s2p1_kernel(const float* __restrict__ x, float* __restrict__ y) {
    const unsigned t = blockIdx.x * BLOCK + threadIdx.x;
    if (t >= TOTAL_THREADS) return;

    const unsigned quad = t % QPR;            // column quad (4 output cols)
    const unsigned rpid = t / QPR;            // global row-pair id = img*PAIRS + p
    const unsigned p    = rpid % PAIRS;       // row pair within image
    const unsigned img  = rpid / PAIRS;

    const float* __restrict__ in = x + (size_t)img * ((size_t)H * W);

    // Input rows for the 2x output rows: 4p-1 .. 4p+3 (row -1 clamped to 0).
    const int r1 = 4 * (int)p;                // always valid (0..220)
    int r0 = r1 - 1;
    if (r0 < 0) r0 = 0;                       // idempotent pad clamp (branchless max)

    const int c0 = 8 * (int)quad;             // first aligned column of the window
    // Left column (col 8q-1); clamp -1 -> 0 for quad 0 (idempotent).
    const int lo = (quad == 0) ? 0 : -1;      // one cndmask, reused for all rows

    const float* rp0 = in + (size_t)r0 * W + c0;
    const float* rp1 = in + (size_t)(r1 + 0) * W + c0;
    const float* rp2 = in + (size_t)(r1 + 1) * W + c0;
    const float* rp3 = in + (size_t)(r1 + 2) * W + c0;
    const float* rp4 = in + (size_t)(r1 + 3) * W + c0;

    // Loads: 5 rows x (b32 + b128 + b128)
    const float l0 = rp0[lo]; const f4 a0 = *(const f4*)rp0; const f4 b0 = *(const f4*)(rp0 + 4);
    const float l1 = rp1[lo]; const f4 a1 = *(const f4*)rp1; const f4 b1 = *(const f4*)(rp1 + 4);
    const float l2 = rp2[lo]; const f4 a2 = *(const f4*)rp2; const f4 b2 = *(const f4*)(rp2 + 4);
    const float l3 = rp3[lo]; const f4 a3 = *(const f4*)rp3; const f4 b3 = *(const f4*)(rp3 + 4);
    const float l4 = rp4[lo]; const f4 a4 = *(const f4*)rp4; const f4 b4 = *(const f4*)(rp4 + 4);

    // Vertical 3-row maxes (middle row r2 shared by both output rows)
    const float lt = fmax3(l0, l1, l2);
    const f4    at = vmax3(a0, a1, a2);
    const f4    bt = vmax3(b0, b1, b2);
    const float lb = fmax3(l2, l3, l4);
    const f4    ab = vmax3(a2, a3, a4);
    const f4    bb = vmax3(b2, b3, b4);

    // Horizontal 3-wide, stride-2 maxes over [l, a.xyzw, b.xyzw]
    f4 ot, ob;
    ot.x = fmax3(lt,   at.x, at.y);
    ot.y = fmax3(at.y, at.z, at.w);
    ot.z = fmax3(at.w, bt.x, bt.y);
    ot.w = fmax3(bt.y, bt.z, bt.w);
    ob.x = fmax3(lb,   ab.x, ab.y);
    ob.y = fmax3(ab.y, ab.z, ab.w);
    ob.z = fmax3(ab.w, bb.x, bb.y);
    ob.w = fmax3(bb.y, bb.z, bb.w);

    // Output rows 2*rpid and 2*rpid+1 (global), never re-read: nontemporal stores
    float* orow = y + (size_t)(2u * rpid) * OW + 4u * quad;
    __builtin_nontemporal_store(ot, (f4*)orow);
    __builtin_nontemporal_store(ob, (f4*)(orow + OW));
}

extern "C" void kernel_launch(void* const* d_in, const int* in_sizes, int n_in,
                              void* d_out, int out_size, void* d_ws, size_t ws_size,
                              hipStream_t stream) {
    const float* x = (const float*)d_in[0];
    float* y = (float*)d_out;
    const unsigned grid = (TOTAL_THREADS + BLOCK - 1) / BLOCK;  // 12544, exact
    maxpool2d_k3s2p1_kernel<<<grid, BLOCK, 0, stream>>>(x, y);
}